// RollerPooling_48034914239013
// MI455X (gfx1250) — compile-verified
//
#include <hip/hip_runtime.h>

// RollerPooling on MI455X (gfx1250, wave32).
//
// out[e,r] = (1/w) * sum_{c=r}^{min(r+w,64)-1} hidden[e,c],  w = 65 - target_size[e]
//
// Strategy: per-edge inclusive prefix sum S[m,c] = sum_{k<=c} H[m,k], computed as
// H[16x64] @ U[64x64] (upper-triangular ones, shared) with V_WMMA_F32_16X16X4_F32
// (fp32 in/accum -> matches fp32 reference precision). Then
// out[r] = (S[min(r+w,64)-1] - (r ? S[r-1] : 0)) * (1/w).
// Kernel is HBM-bound (~52 MB total -> ~2.2 us at 23.3 TB/s); WMMA prefix keeps
// VALU work negligible. pool_matrix input is never read (window is analytic).

typedef __attribute__((ext_vector_type(2))) float v2f;
typedef __attribute__((ext_vector_type(4))) float v4f;
typedef __attribute__((ext_vector_type(8))) float v8f;

#define NCH   64
#define TILE  16          // edges per wave
#define SSTR  68          // LDS row stride in floats (pad vs bank conflicts)
#define WPB   8           // waves per 256-thread block

__global__ __launch_bounds__(256) void roller_pool_wmma(
    const float* __restrict__ hidden,   // [E, 64] fp32
    const int*   __restrict__ tsize,    // [E] (1..64)
    float*       __restrict__ out,      // [E, 64] fp32
    int n_edges)
{
  __shared__ float lds[WPB][TILE * SSTR];

  const int lane = threadIdx.x & 31;
  const int wv   = threadIdx.x >> 5;
  const int base = (blockIdx.x * WPB + wv) * TILE;   // first edge of this wave's tile
  const bool active = base < n_edges;                 // wave-uniform -> EXEC all-1s inside

  const int lo = lane & 15;   // N / M index within 16
  const int hi = lane >> 4;   // half-wave select (K offset)
  float* S = &lds[wv][0];

  if (active) {
    int row = base + lo;
    if (row >= n_edges) row = n_edges - 1;            // clamp reads on ragged tail

#if __has_builtin(__builtin_amdgcn_wmma_f32_16x16x4_f32)
    // A-matrix (16x4 f32) layout: lane = M%16; VGPR0/1 hold K = 4kc+2*hi, +1.
    const float* hrow = hidden + (size_t)row * NCH + 2 * hi;
    v2f a[16];
#pragma unroll
    for (int kc = 0; kc < 16; ++kc)
      a[kc] = *(const v2f*)(hrow + 4 * kc);           // tile read exactly once (global_load_b64)

#pragma unroll
    for (int nt = 0; nt < 4; ++nt) {                  // 16-wide column tiles of S
      v8f acc = {};
      const int cend = 16 * nt + lo;                  // this lane's global column
#pragma unroll
      for (int kc = 0; kc < 4 * nt + 4; ++kc) {       // skip all-zero chunks above diagonal
        const int kb = 4 * kc + 2 * hi;
        // B-matrix (4x16 f32): lane = N%16; VGPR0/1 hold K = 2*hi, 2*hi+1 of chunk.
        v2f b;
        b.x = (kb     <= cend) ? 1.0f : 0.0f;         // U[k,c] = (k <= c)
        b.y = (kb + 1 <= cend) ? 1.0f : 0.0f;
        acc = __builtin_amdgcn_wmma_f32_16x16x4_f32(
            /*neg_a=*/false, a[kc], /*neg_b=*/false, b,
            /*c_mod=*/(short)0, acc, /*reuse_a=*/false, /*reuse_b=*/false);
      }
      // D layout: VGPR j holds M = j + 8*hi, N = lo  -> spill S tile to LDS
#pragma unroll
      for (int j = 0; j < 8; ++j)
        S[(j + 8 * hi) * SSTR + 16 * nt + lo] = acc[j];
    }
#else
    // Fallback (should not trigger on gfx1250): serial prefix per edge.
    if (hi == 0) {
      const float* hr = hidden + (size_t)row * NCH;
      float run = 0.0f;
      for (int k = 0; k < NCH; ++k) { run += hr[k]; S[lo * SSTR + k] = run; }
    }
#endif
  }

  __syncthreads();  // make cross-lane LDS S-tile visible (compiler ordering + waits)

  if (active) {
    const int m  = lane >> 1;                 // edge within tile (2 lanes per edge)
    const int r0 = (lane & 1) * 32;           // each lane covers 32 output channels
    const int e  = base + m;
    if (e < n_edges) {
      const int   w    = 65 - tsize[e];
      const float invw = 1.0f / (float)w;
      const float* Sm  = S + m * SSTR;
      float* orow = out + (size_t)e * NCH;
#pragma unroll
      for (int g = 0; g < 8; ++g) {
        const int rb = r0 + 4 * g;
        v4f o;
#pragma unroll
        for (int q = 0; q < 4; ++q) {
          const int r   = rb + q;
          int       top = r + w; if (top > NCH) top = NCH;
          const float hp = Sm[top - 1];
          const float lp = (r == 0) ? 0.0f : Sm[r - 1];
          o[q] = (hp - lp) * invw;
        }
        *(v4f*)(orow + rb) = o;               // coalesced global_store_b128
      }
    }
  }
}

extern "C" void kernel_launch(void* const* d_in, const int* in_sizes, int n_in,
                              void* d_out, int out_size, void* d_ws, size_t ws_size,
                              hipStream_t stream) {
  (void)n_in; (void)d_ws; (void)ws_size; (void)out_size;
  const float* hidden = (const float*)d_in[0];
  const int*   tsize  = (const int*)d_in[1];   // harness delivers integer inputs as int32
  // d_in[2] (pool_matrix) is unused: the pooling window is computed analytically.
  float* out = (float*)d_out;

  const int n_edges = in_sizes[1];             // [E]
  const int tiles   = (n_edges + TILE - 1) / TILE;
  const int blocks  = (tiles + WPB - 1) / WPB;
  roller_pool_wmma<<<blocks, 256, 0, stream>>>(hidden, tsize, out, n_edges);
}